// GraphNet_56435870269624
// MI455X (gfx1250) — compile-verified
//
#include <hip/hip_runtime.h>
#include <hip/hip_bf16.h>
#include <math.h>

typedef __attribute__((ext_vector_type(2))) float v2f;
typedef __attribute__((ext_vector_type(8))) float v8f;

#define NNODES 16384
#define NGENES 12132
#define NG     64          // graphs == output columns
#define NEDGE  262144
#define KC     64          // K-chunk staged in LDS
#define KSTEPS (KC / 4)    // 16 WMMA K-steps per chunk
#define CHUNKF (KC * NG)   // 4096 floats per chunk

// ---------------------------------------------------------------------------
// 0) Pre-scale weights: Wp[j,g] = gcn_weight[j,g] / node_scales[j]
// ---------------------------------------------------------------------------
__global__ __launch_bounds__(256) void prescale_kernel(
    const float* __restrict__ gcn_weight, const float* __restrict__ node_scales,
    float* __restrict__ Wp) {
  int idx = blockIdx.x * 256 + threadIdx.x;
  if (idx < NGENES * NG) {
    int j = idx >> 6;  // gene row (NG == 64)
    Wp[idx] = gcn_weight[idx] * __frcp_rn(node_scales[j]);
  }
}

// swizzled LDS index for B element (k in 0..KC-1, col in 0..63):
// fragment layout so each lane's (K,K+1) pair per tile is one 8B ds_load_b64
__device__ __forceinline__ int bswz(int k, int col) {
  return ((k >> 2) << 8) + (((k >> 1) & 1) << 7) + (col << 1) + (k & 1);
}

// ---------------------------------------------------------------------------
// 1) Main GEMM: xw[N,64] = x[N,12132] @ Wp[12132,64] via V_WMMA_F32_16X16X4_F32
//    One wave -> 16 rows x all 64 columns (4 accumulator tiles).
//    Double-buffered, pre-swizzled B in LDS; A fragments preloaded per chunk.
// ---------------------------------------------------------------------------
__global__ __launch_bounds__(256) void gemm_kernel(
    const float* __restrict__ x, const float* __restrict__ Wp,
    float* __restrict__ xw) {
  __shared__ __align__(16) float ldsB[2][CHUNKF];  // 32 KB

  const int tid   = threadIdx.x;
  const int lane  = tid & 31;
  const int wave  = blockIdx.x * 8 + (tid >> 5);   // 0..1023
  const int rowB  = wave * 16;                     // 16-row stripe of x

  const int mrow  = lane & 15;                     // A: M index
  const int khalf = (lane >> 4) << 1;              // A: K sub-offset 0 or 2
  const int ncol  = lane & 15;                     // B/C: N index
  const int mAdd  = (lane >> 4) << 3;              // C: +8 rows for hi lanes
  // lane's B fragment base inside a swizzled chunk (floats)
  const int bBase = ((lane >> 4) << 7) + (ncol << 1);

  v8f c0 = {}, c1 = {}, c2 = {}, c3 = {};

  const float* __restrict__ xrow = x + (size_t)(rowB + mrow) * NGENES + khalf;

  float stage[16];

  // ---- preload + store chunk 0 (full) ----
#pragma unroll
  for (int j = 0; j < 16; ++j) stage[j] = Wp[tid + j * 256];
#pragma unroll
  for (int j = 0; j < 16; ++j) {
    int idx = tid + j * 256;
    ldsB[0][bswz(idx >> 6, idx & 63)] = stage[j];
  }
  __syncthreads();

  const int nFull = NGENES / KC;  // 189 full chunks; remainder = 36
  for (int c = 0; c < nFull; ++c) {
    const int buf = c & 1;
    const int k0  = c * KC;

    // ---- issue next B chunk's global loads early ----
    {
      const int k1  = k0 + KC;
      const int lim = (NGENES - k1 < KC ? NGENES - k1 : KC) * NG;
      const float* src = Wp + (size_t)k1 * NG;
      if (lim == CHUNKF) {
#pragma unroll
        for (int j = 0; j < 16; ++j) stage[j] = src[tid + j * 256];
      } else {
#pragma unroll
        for (int j = 0; j < 16; ++j) {
          int idx = tid + j * 256;
          stage[j] = (idx < lim) ? src[idx] : 0.f;
        }
      }
    }

    // ---- preload all A fragments for this chunk (16 loads in flight) ----
    __builtin_prefetch(xrow + k0 + KC, 0, 1);
    v2f aF[KSTEPS];
#pragma unroll
    for (int s = 0; s < KSTEPS; ++s)
      aF[s] = *(const v2f*)(xrow + k0 + 4 * s);

    // ---- compute: 64 WMMAs, B via single ds_load_b64 per tile ----
    const float* bc = &ldsB[buf][bBase];
#pragma unroll
    for (int s = 0; s < KSTEPS; ++s) {
      const float* bp = bc + (s << 8);
      v2f b0 = *(const v2f*)(bp);
      v2f b1 = *(const v2f*)(bp + 32);
      v2f b2 = *(const v2f*)(bp + 64);
      v2f b3 = *(const v2f*)(bp + 96);
      c0 = __builtin_amdgcn_wmma_f32_16x16x4_f32(false, aF[s], false, b0, (short)0, c0, false, false);
      c1 = __builtin_amdgcn_wmma_f32_16x16x4_f32(false, aF[s], false, b1, (short)0, c1, false, false);
      c2 = __builtin_amdgcn_wmma_f32_16x16x4_f32(false, aF[s], false, b2, (short)0, c2, false, false);
      c3 = __builtin_amdgcn_wmma_f32_16x16x4_f32(false, aF[s], false, b3, (short)0, c3, false, false);
    }

    // ---- commit staged chunk to the other buffer ----
#pragma unroll
    for (int j = 0; j < 16; ++j) {
      int idx = tid + j * 256;
      ldsB[buf ^ 1][bswz(idx >> 6, idx & 63)] = stage[j];
    }
    __syncthreads();
  }

  // ---- epilogue: remainder chunk (kc = 36 -> 9 K-steps) ----
  {
    const int buf = nFull & 1;
    const int k0  = nFull * KC;       // 12096
    const int kc  = NGENES - k0;      // 36
    const float* bc = &ldsB[buf][bBase];
    for (int kk = 0; kk < kc; kk += 4) {
      v2f a = *(const v2f*)(xrow + k0 + kk);
      const float* bp = bc + ((kk >> 2) << 8);
      v2f b0 = *(const v2f*)(bp);
      v2f b1 = *(const v2f*)(bp + 32);
      v2f b2 = *(const v2f*)(bp + 64);
      v2f b3 = *(const v2f*)(bp + 96);
      c0 = __builtin_amdgcn_wmma_f32_16x16x4_f32(false, a, false, b0, (short)0, c0, false, false);
      c1 = __builtin_amdgcn_wmma_f32_16x16x4_f32(false, a, false, b1, (short)0, c1, false, false);
      c2 = __builtin_amdgcn_wmma_f32_16x16x4_f32(false, a, false, b2, (short)0, c2, false, false);
      c3 = __builtin_amdgcn_wmma_f32_16x16x4_f32(false, a, false, b3, (short)0, c3, false, false);
    }
  }

  // ---- store C tiles: VGPR v -> row (rowB + mAdd + v), col (tile*16 + ncol)
#pragma unroll
  for (int v = 0; v < 8; ++v) {
    float* orow = xw + (size_t)(rowB + mAdd + v) * NG;
    orow[ncol]      = c0[v];
    orow[16 + ncol] = c1[v];
    orow[32 + ncol] = c2[v];
    orow[48 + ncol] = c3[v];
  }
}

// ---------------------------------------------------------------------------
// 2) agg = xw + gcn_bias  (self-loop term + bias, before edge accumulation)
// ---------------------------------------------------------------------------
__global__ __launch_bounds__(256) void init_agg_kernel(
    const float* __restrict__ xw, const float* __restrict__ gcn_bias,
    float* __restrict__ agg) {
  int idx = blockIdx.x * 256 + threadIdx.x;
  if (idx < NNODES * NG) agg[idx] = xw[idx] + gcn_bias[idx & 63];
}

// ---------------------------------------------------------------------------
// 3) Edge scatter: agg[dst] += xw[src]   (4 MB target, L2-resident atomics)
// ---------------------------------------------------------------------------
__global__ __launch_bounds__(256) void edge_kernel(
    const int* __restrict__ src, const int* __restrict__ dst,
    const float* __restrict__ xw, float* __restrict__ agg) {
  int t = threadIdx.x;
  int e = blockIdx.x * 4 + (t >> 6);
  int g = t & 63;
  int s = src[e];
  int d = dst[e];
  atomicAdd(&agg[(size_t)d * NG + g], xw[(size_t)s * NG + g]);
}

// ---------------------------------------------------------------------------
// 4) selected[i] = agg[i, graph_ids[i]]
// ---------------------------------------------------------------------------
__global__ __launch_bounds__(256) void select_kernel(
    const float* __restrict__ agg, const int* __restrict__ graph_ids,
    float* __restrict__ sel) {
  int i = blockIdx.x * 256 + threadIdx.x;
  if (i < NNODES) sel[i] = agg[(size_t)i * NG + graph_ids[i]];
}

// ---------------------------------------------------------------------------
// 5) h[g] = ((w1[g,:] . selected) + b1[g] - final_means[g]) / final_scales[g]
// ---------------------------------------------------------------------------
__global__ __launch_bounds__(256) void h_kernel(
    const float* __restrict__ w1, const float* __restrict__ sel,
    const float* __restrict__ b1, const float* __restrict__ fm,
    const float* __restrict__ fs, float* __restrict__ hn) {
  __shared__ float red[256];
  const int g = blockIdx.x;
  const float* wrow = w1 + (size_t)g * NNODES;
  float s = 0.f;
  for (int i = threadIdx.x; i < NNODES; i += 256) s += wrow[i] * sel[i];
  red[threadIdx.x] = s;
  __syncthreads();
  for (int off = 128; off > 0; off >>= 1) {
    if (threadIdx.x < off) red[threadIdx.x] += red[threadIdx.x + off];
    __syncthreads();
  }
  if (threadIdx.x == 0) hn[g] = (red[0] + b1[g] - fm[g]) * __frcp_rn(fs[g]);
}

// ---------------------------------------------------------------------------
// 6) out = sigmoid(w2 . h + b2)
// ---------------------------------------------------------------------------
__global__ __launch_bounds__(64) void final_kernel(
    const float* __restrict__ w2, const float* __restrict__ hn,
    const float* __restrict__ b2, float* __restrict__ out) {
  __shared__ float red[64];
  int t = threadIdx.x;
  red[t] = w2[t] * hn[t];
  __syncthreads();
  for (int off = 32; off > 0; off >>= 1) {
    if (t < off) red[t] += red[t + off];
    __syncthreads();
  }
  if (t == 0) out[0] = 1.f / (1.f + __expf(-(red[0] + b2[0])));
}

// ---------------------------------------------------------------------------
extern "C" void kernel_launch(void* const* d_in, const int* in_sizes, int n_in,
                              void* d_out, int out_size, void* d_ws, size_t ws_size,
                              hipStream_t stream) {
  (void)in_sizes; (void)n_in; (void)out_size; (void)ws_size;

  const float* x           = (const float*)d_in[0];
  const int*   edge_index  = (const int*)  d_in[1];   // [2, E]
  const int*   graph_ids   = (const int*)  d_in[2];
  const float* node_scales = (const float*)d_in[3];
  const float* gcn_weight  = (const float*)d_in[4];
  const float* gcn_bias    = (const float*)d_in[5];
  const float* w1          = (const float*)d_in[6];
  const float* b1          = (const float*)d_in[7];
  const float* final_means = (const float*)d_in[8];
  const float* final_scales= (const float*)d_in[9];
  const float* w2          = (const float*)d_in[10];
  const float* b2          = (const float*)d_in[11];
  float* out = (float*)d_out;

  // workspace carve-up (floats)
  float* ws  = (float*)d_ws;
  float* Wp  = ws;                                  // NGENES*NG   = 776448
  float* xw  = Wp  + (size_t)NGENES * NG;           // NNODES*NG   = 1048576
  float* agg = xw  + (size_t)NNODES * NG;           // NNODES*NG   = 1048576
  float* sel = agg + (size_t)NNODES * NG;           // NNODES      = 16384
  float* hn  = sel + NNODES;                        // NG          = 64

  const int* esrc = edge_index;
  const int* edst = edge_index + NEDGE;

  prescale_kernel<<<(NGENES * NG + 255) / 256, 256, 0, stream>>>(gcn_weight, node_scales, Wp);
  gemm_kernel<<<NNODES / (16 * 8), 256, 0, stream>>>(x, Wp, xw);          // 128 blocks
  init_agg_kernel<<<(NNODES * NG + 255) / 256, 256, 0, stream>>>(xw, gcn_bias, agg);
  edge_kernel<<<NEDGE / 4, 256, 0, stream>>>(esrc, edst, xw, agg);
  select_kernel<<<(NNODES + 255) / 256, 256, 0, stream>>>(agg, graph_ids, sel);
  h_kernel<<<NG, 256, 0, stream>>>(w1, sel, b1, final_means, final_scales, hn);
  final_kernel<<<1, 64, 0, stream>>>(w2, hn, b2, out);
}